// AggregatorL1_69475390980334
// MI455X (gfx1250) — compile-verified
//
#include <hip/hip_runtime.h>
#include <hip/hip_bf16.h>

typedef __attribute__((ext_vector_type(2))) float v2f;
typedef __attribute__((ext_vector_type(4))) float v4f;
typedef __attribute__((ext_vector_type(8))) float v8f;

#define FVLEN 256
#define HDIM 256
#define NNEIGH 32
#define MTILE 64
#define LDS_STRIDE 260  // 256 + 4 floats pad -> conflict-free 16x16x4 fragment reads

__global__ __launch_bounds__(256)
void sage_fused_kernel(const int* __restrict__ x0,
                       const int* __restrict__ x1,
                       const float* __restrict__ emb,
                       const float* __restrict__ W_self,
                       const float* __restrict__ b_self,
                       const float* __restrict__ W_neigh,
                       const float* __restrict__ b_neigh,
                       float* __restrict__ out)
{
    extern __shared__ float lds[];
    float* ldsS = lds;                       // self features  [64][260]
    float* ldsN = lds + MTILE * LDS_STRIDE;  // pooled features[64][260]

    const int lane = threadIdx.x & 31;
    const int wv   = threadIdx.x >> 5;       // wave id 0..7 (wave32)
    const int rowbase = blockIdx.x * MTILE;

    // ---------------- Phase A: gather self rows + mean-pool neighbors into LDS
    // 8 waves x 8 rows; each lane covers 8 consecutive floats of the 256-f row.
    {
        const int fo = lane * 8;
        for (int r = 0; r < 8; ++r) {
            const int lrow = wv * 8 + r;
            const int b    = rowbase + lrow;

            // self feature row
            const size_t si = (size_t)x0[b] * FVLEN;
            v4f s0 = *(const v4f*)(emb + si + fo);
            v4f s1 = *(const v4f*)(emb + si + fo + 4);
            *(v4f*)(ldsS + (size_t)lrow * LDS_STRIDE + fo)     = s0;
            *(v4f*)(ldsS + (size_t)lrow * LDS_STRIDE + fo + 4) = s1;

            // neighbor mean pool
            v4f a0 = (v4f){0.f, 0.f, 0.f, 0.f};
            v4f a1 = (v4f){0.f, 0.f, 0.f, 0.f};
            const int* nb = x1 + (size_t)b * NNEIGH;
            #pragma unroll 4
            for (int j = 0; j < NNEIGH; ++j) {
                const size_t ni = (size_t)nb[j] * FVLEN;
                a0 += *(const v4f*)(emb + ni + fo);
                a1 += *(const v4f*)(emb + ni + fo + 4);
            }
            const float inv = 1.0f / (float)NNEIGH;
            a0 *= inv;
            a1 *= inv;
            *(v4f*)(ldsN + (size_t)lrow * LDS_STRIDE + fo)     = a0;
            *(v4f*)(ldsN + (size_t)lrow * LDS_STRIDE + fo + 4) = a1;
        }
    }
    __syncthreads();

    // ---------------- Phase B: WMMA f32 GEMM. Each wave: 64 rows x 64 cols.
    // Output cols 0..255 = self GEMM, 256..511 = neigh GEMM.
    const int colbase = wv * 64;
    const float* Alds;
    const float* Wmat;
    const float* bias;
    int wcol;
    if (colbase < HDIM) { Alds = ldsS; Wmat = W_self;  bias = b_self;  wcol = colbase; }
    else                { Alds = ldsN; Wmat = W_neigh; bias = b_neigh; wcol = colbase - HDIM; }

    const int mrow  = lane & 15;        // row (A) / col-of-W (B) within 16-subtile
    const int khalf = (lane >> 4) * 2;  // K offset carried by upper half-wave

    // init accumulators with bias (D/C layout: n = lane&15 for all 8 VGPRs)
    v8f acc[4][4];
    #pragma unroll
    for (int ni = 0; ni < 4; ++ni) {
        const float bv = bias[wcol + ni * 16 + mrow];
        #pragma unroll
        for (int mi = 0; mi < 4; ++mi)
            acc[mi][ni] = (v8f){bv, bv, bv, bv, bv, bv, bv, bv};
    }

    for (int k = 0; k < FVLEN; k += 4) {
        v2f af[4], bf[4];
        #pragma unroll
        for (int mi = 0; mi < 4; ++mi)
            af[mi] = *(const v2f*)(Alds + (size_t)(mi * 16 + mrow) * LDS_STRIDE + k + khalf);
        #pragma unroll
        for (int ni = 0; ni < 4; ++ni)
            bf[ni] = *(const v2f*)(Wmat + (size_t)(wcol + ni * 16 + mrow) * FVLEN + k + khalf);
        #pragma unroll
        for (int mi = 0; mi < 4; ++mi) {
            #pragma unroll
            for (int ni = 0; ni < 4; ++ni) {
                acc[mi][ni] = __builtin_amdgcn_wmma_f32_16x16x4_f32(
                    false, af[mi], false, bf[ni], (short)0, acc[mi][ni], false, false);
            }
        }
    }

    // ---------------- store with ReLU
    const int mhi = (lane >> 4) * 8;    // D rows: VGPR v -> m = v + 8*(lane>=16)
    #pragma unroll
    for (int mi = 0; mi < 4; ++mi) {
        #pragma unroll
        for (int ni = 0; ni < 4; ++ni) {
            const int col = colbase + ni * 16 + mrow;
            #pragma unroll
            for (int v = 0; v < 8; ++v) {
                const int m = rowbase + mi * 16 + mhi + v;
                float x = acc[mi][ni][v];
                out[(size_t)m * (2 * HDIM) + col] = fmaxf(x, 0.0f);
            }
        }
    }
}

extern "C" void kernel_launch(void* const* d_in, const int* in_sizes, int n_in,
                              void* d_out, int out_size, void* d_ws, size_t ws_size,
                              hipStream_t stream) {
    const int*   x0      = (const int*)d_in[0];
    const int*   x1      = (const int*)d_in[1];
    const float* emb     = (const float*)d_in[2];
    const float* W_self  = (const float*)d_in[3];
    const float* b_self  = (const float*)d_in[4];
    const float* W_neigh = (const float*)d_in[5];
    const float* b_neigh = (const float*)d_in[6];
    float* out = (float*)d_out;

    const int B    = in_sizes[0];      // 16384
    const int grid = B / MTILE;        // 256 workgroups
    const size_t shmem = (size_t)2 * MTILE * LDS_STRIDE * sizeof(float); // 133120 B

    sage_fused_kernel<<<grid, 256, shmem, stream>>>(
        x0, x1, emb, W_self, b_self, W_neigh, b_neigh, out);
}